// FGAT_67173288509774
// MI455X (gfx1250) — compile-verified
//
#include <hip/hip_runtime.h>
#include <hip/hip_bf16.h>
#include <math.h>

#define NN 50000
#define NE 1600000
#define NP 4096
#define LAYERS 4
#define FW 192   // fused node-feature width: [f_ni(32) | f_nj(32) | hh(128)]

typedef __attribute__((ext_vector_type(2))) float v2f;
typedef __attribute__((ext_vector_type(8))) float v8f;

// Order-preserving float <-> uint encoding for atomicMax-based segment_max.
__device__ __forceinline__ unsigned fenc(float f) {
  unsigned u = __float_as_uint(f);
  return (u & 0x80000000u) ? ~u : (u | 0x80000000u);
}
__device__ __forceinline__ float fdec(unsigned u) {
  unsigned v = (u & 0x80000000u) ? (u & 0x7FFFFFFFu) : ~u;
  return __uint_as_float(v);
}

// ---------------- per-layer weight concat: Wcat (32 x 192) = [W_ni | W_nj | W_node]
__global__ void prep_wcat(const float* __restrict__ Wni, const float* __restrict__ Wnj,
                          const float* __restrict__ Wnode, float* __restrict__ wcat, int l) {
  int g = blockIdx.x * blockDim.x + threadIdx.x;
  if (g >= 32 * FW) return;
  int k = g / FW, c = g % FW;
  float v;
  if (c < 32)       v = Wni[(size_t)l * 1024 + k * 32 + c];
  else if (c < 64)  v = Wnj[(size_t)l * 1024 + k * 32 + (c - 32)];
  else              v = Wnode[(size_t)l * 4096 + k * 128 + (c - 64)];
  wcat[g] = v;
}

// ---------------- zero agg/denom, reset mx (encoded -inf == 0)
__global__ void init_layer(float* __restrict__ agg, float* __restrict__ denom,
                           unsigned* __restrict__ mx) {
  int g = blockIdx.x * blockDim.x + threadIdx.x;
  if (g < NN * 128) agg[g] = 0.f;
  if (g < NN * 4) { denom[g] = 0.f; mx[g] = 0u; }
}

// ---------------- WMMA node GEMM: F(50000x192) = h(50000x32) @ Wcat(32x192)
// one wave per 16x16 output tile, K=32 via 8x v_wmma_f32_16x16x4_f32 (exact f32).
// Result stored contiguously (row stride 192): branch-free coalesced stores.
__global__ void __launch_bounds__(256) node_matmul(
    const float* __restrict__ h, const float* __restrict__ wcat, float* __restrict__ F) {
  const int ROWT = NN / 16;   // 3125 (exact)
  const int COLT = FW / 16;   // 12
  int wid = blockIdx.x * (blockDim.x >> 5) + (threadIdx.x >> 5);
  if (wid >= ROWT * COLT) return;  // whole-wave exit; EXEC all-1s at WMMA
  int rt = wid / COLT, ct = wid % COLT;
  int lane = threadIdx.x & 31;
  int half = lane >> 4, l16 = lane & 15;
  const float* arow = h + (size_t)(rt * 16 + l16) * 32;
  int col = ct * 16 + l16;
  v8f acc = {0.f, 0.f, 0.f, 0.f, 0.f, 0.f, 0.f, 0.f};
#pragma unroll
  for (int kk = 0; kk < 8; ++kk) {
    int kb = kk * 4 + half * 2;  // A: lanes0-15 K={0,1}; lanes16-31 K={2,3} (ISA layout)
    v2f a; a.x = arow[kb];                        a.y = arow[kb + 1];
    v2f b; b.x = wcat[(size_t)kb * FW + col];     b.y = wcat[(size_t)(kb + 1) * FW + col];
    acc = __builtin_amdgcn_wmma_f32_16x16x4_f32(false, a, false, b, (short)0, acc, false, false);
  }
  float* outp = F + (size_t)(rt * 16 + half * 8) * FW + col;
#pragma unroll
  for (int r = 0; r < 8; ++r) {  // D: VGPR r -> row r (lanes 0-15) / r+8 (lanes 16-31)
    outp[(size_t)r * FW] = acc[r];
  }
}

// ---------------- per-edge: fe = e@W_fij + b, f = F[src,0:32]+F[dst,32:64]+fe,
// leaky_relu, per-head logits, atomic segment-max over dst. All gathers b128.
__global__ void __launch_bounds__(256) edge_logits(
    const float* __restrict__ efeat, const int* __restrict__ src, const int* __restrict__ dst,
    const float* __restrict__ F,
    const float* __restrict__ Wf, const float* __restrict__ bE, const float* __restrict__ attn,
    float* __restrict__ zlog, unsigned* __restrict__ mx) {
  __shared__ float sWf[256], sB[32], sA[32];
  for (int i = threadIdx.x; i < 256; i += blockDim.x) sWf[i] = Wf[i];
  if (threadIdx.x < 32) { sB[threadIdx.x] = bE[threadIdx.x]; sA[threadIdx.x] = attn[threadIdx.x]; }
  __syncthreads();
  int e = blockIdx.x * blockDim.x + threadIdx.x;
  if (e >= NE) return;
  int s = src[e], d = dst[e];
  const float4* ef4 = (const float4*)(efeat + (size_t)e * 8);
  float4 e0 = ef4[0], e1 = ef4[1];
  float ef[8] = {e0.x, e0.y, e0.z, e0.w, e1.x, e1.y, e1.z, e1.w};
  const float4* fs4 = (const float4*)(F + (size_t)s * FW);        // f_ni[src]
  const float4* fd4 = (const float4*)(F + (size_t)d * FW + 32);   // f_nj[dst]
  float fs[32], fd[32];
#pragma unroll
  for (int q = 0; q < 8; ++q) {
    float4 a = fs4[q], b = fd4[q];
    fs[q * 4 + 0] = a.x; fs[q * 4 + 1] = a.y; fs[q * 4 + 2] = a.z; fs[q * 4 + 3] = a.w;
    fd[q * 4 + 0] = b.x; fd[q * 4 + 1] = b.y; fd[q * 4 + 2] = b.z; fd[q * 4 + 3] = b.w;
  }
  float lg[4] = {0.f, 0.f, 0.f, 0.f};
#pragma unroll
  for (int c = 0; c < 32; ++c) {
    float fe = sB[c];
#pragma unroll
    for (int k = 0; k < 8; ++k) fe += ef[k] * sWf[k * 32 + c];
    float f = fs[c] + fd[c] + fe;
    f = f > 0.f ? f : 0.01f * f;        // jax.nn.leaky_relu default slope
    lg[c >> 3] += f * sA[c];            // attn[h][d] flattened == index c
  }
  *(float4*)(zlog + (size_t)e * 4) = make_float4(lg[0], lg[1], lg[2], lg[3]);
#pragma unroll
  for (int hd = 0; hd < 4; ++hd) atomicMax(&mx[(size_t)d * 4 + hd], fenc(lg[hd]));
}

// ---------------- z = exp(logit - mx[dst]); denom[dst] += z  (z overwrites logits)
__global__ void edge_denom(const int* __restrict__ dst, float* __restrict__ zlog,
                           const unsigned* __restrict__ mx, float* __restrict__ denom) {
  int e = blockIdx.x * blockDim.x + threadIdx.x;
  if (e >= NE) return;
  int d = dst[e];
  float4 lg = *(const float4*)(zlog + (size_t)e * 4);
  float z0 = expf(lg.x - fdec(mx[(size_t)d * 4 + 0]));
  float z1 = expf(lg.y - fdec(mx[(size_t)d * 4 + 1]));
  float z2 = expf(lg.z - fdec(mx[(size_t)d * 4 + 2]));
  float z3 = expf(lg.w - fdec(mx[(size_t)d * 4 + 3]));
  *(float4*)(zlog + (size_t)e * 4) = make_float4(z0, z1, z2, z3);
  atomicAdd(&denom[(size_t)d * 4 + 0], z0);
  atomicAdd(&denom[(size_t)d * 4 + 1], z1);
  atomicAdd(&denom[(size_t)d * 4 + 2], z2);
  atomicAdd(&denom[(size_t)d * 4 + 3], z3);
}

// ---------------- agg[dst] += hh[src] * (z/denom[dst]); one thread per (edge, 4 channels)
__global__ void edge_agg(const int* __restrict__ src, const int* __restrict__ dst,
                         const float* __restrict__ F, const float* __restrict__ zlog,
                         const float* __restrict__ denom, float* __restrict__ agg) {
  size_t g = (size_t)blockIdx.x * blockDim.x + threadIdx.x;
  if (g >= (size_t)NE * 32) return;
  size_t e = g >> 5;
  int c4 = (int)(g & 31) * 4;       // channel group base (0..124), head = c4>>5
  int hd = c4 >> 5;
  int s = src[e], d = dst[e];
  float a = zlog[e * 4 + hd] / denom[(size_t)d * 4 + hd];
  float4 hv = *(const float4*)(F + (size_t)s * FW + 64 + c4);   // hh[src]
  float* ap = agg + (size_t)d * 128 + c4;
  atomicAdd(ap + 0, hv.x * a);
  atomicAdd(ap + 1, hv.y * a);
  atomicAdd(ap + 2, hv.z * a);
  atomicAdd(ap + 3, hv.w * a);
}

// ---------------- h = elu(sum over heads); append to states
__global__ void node_update(const float* __restrict__ agg, float* __restrict__ hcur,
                            float* __restrict__ states, int l) {
  int g = blockIdx.x * blockDim.x + threadIdx.x;
  if (g >= NN * 32) return;
  int n = g >> 5, j = g & 31;
  const float* an = agg + (size_t)n * 128;
  float v = an[j] + an[32 + j] + an[64 + j] + an[96 + j];
  v = v > 0.f ? v : expm1f(v);  // elu alpha=1
  hcur[(size_t)n * 32 + j] = v;
  states[(size_t)n * 128 + l * 32 + j] = v;
}

// ---------------- pair MLP: 256 -> relu 128 -> sigmoid 1; one block (128 thr) per pair
__global__ void __launch_bounds__(128) pair_mlp(
    const float* __restrict__ states, const int* __restrict__ users, const int* __restrict__ items,
    const float* __restrict__ W1, const float* __restrict__ b1,
    const float* __restrict__ W2, const float* __restrict__ b2, float* __restrict__ out) {
  __shared__ float z[256];
  __shared__ float red[128];
  int p = blockIdx.x, t = threadIdx.x;
  int u = users[p], it = items[p];
  z[t]       = states[(size_t)u  * 128 + t];
  z[128 + t] = states[(size_t)it * 128 + t];
  __syncthreads();
  float acc = b1[t];
#pragma unroll 8
  for (int k = 0; k < 256; ++k) acc += z[k] * W1[(size_t)k * 128 + t];
  acc = acc > 0.f ? acc : 0.f;
  red[t] = acc * W2[t];
  __syncthreads();
  for (int s = 64; s > 0; s >>= 1) { if (t < s) red[t] += red[t + s]; __syncthreads(); }
  if (t == 0) out[p] = 1.f / (1.f + expf(-(red[0] + b2[0])));
}

extern "C" void kernel_launch(void* const* d_in, const int* in_sizes, int n_in,
                              void* d_out, int out_size, void* d_ws, size_t ws_size,
                              hipStream_t stream) {
  const float* x      = (const float*)d_in[0];
  const float* e      = (const float*)d_in[1];
  const float* W_ni   = (const float*)d_in[2];
  const float* W_nj   = (const float*)d_in[3];
  const float* W_fij  = (const float*)d_in[4];
  const float* b_edge = (const float*)d_in[5];
  const float* attn   = (const float*)d_in[6];
  const float* W_node = (const float*)d_in[7];
  const float* W1     = (const float*)d_in[8];
  const float* b1     = (const float*)d_in[9];
  const float* W2     = (const float*)d_in[10];
  const float* b2     = (const float*)d_in[11];
  const int*   src    = (const int*)d_in[12];
  const int*   dst    = (const int*)d_in[13];
  const int*   users  = (const int*)d_in[14];
  const int*   items  = (const int*)d_in[15];
  float* out = (float*)d_out;

  // workspace layout (floats)
  float* W = (float*)d_ws;
  size_t o = 0;
  float*    hcur   = W + o; o += (size_t)NN * 32;
  float*    F      = W + o; o += (size_t)NN * FW;     // [f_ni | f_nj | hh] per node
  float*    agg    = W + o; o += (size_t)NN * 128;
  float*    states = W + o; o += (size_t)NN * 128;
  float*    denom  = W + o; o += (size_t)NN * 4;
  unsigned* mx     = (unsigned*)(W + o); o += (size_t)NN * 4;
  float*    zlog   = W + o; o += (size_t)NE * 4;
  float*    wcat   = W + o; o += (size_t)32 * FW;

  const int WAVES = (NN / 16) * (FW / 16);          // 37500 wmma tiles
  const int MM_BLOCKS = (WAVES + 7) / 8;            // 8 waves / 256-thr block

  for (int l = 0; l < LAYERS; ++l) {
    const float* h = (l == 0) ? x : hcur;
    prep_wcat<<<(32 * FW + 255) / 256, 256, 0, stream>>>(W_ni, W_nj, W_node, wcat, l);
    init_layer<<<(NN * 128 + 255) / 256, 256, 0, stream>>>(agg, denom, mx);
    node_matmul<<<MM_BLOCKS, 256, 0, stream>>>(h, wcat, F);
    edge_logits<<<(NE + 255) / 256, 256, 0, stream>>>(
        e, src, dst, F, W_fij + (size_t)l * 256, b_edge + (size_t)l * 32,
        attn + (size_t)l * 32, zlog, mx);
    edge_denom<<<(NE + 255) / 256, 256, 0, stream>>>(dst, zlog, mx, denom);
    edge_agg<<<(int)(((long long)NE * 32 + 255) / 256), 256, 0, stream>>>(
        src, dst, F, zlog, denom, agg);
    node_update<<<(NN * 32 + 255) / 256, 256, 0, stream>>>(agg, hcur, states, l);
  }
  pair_mlp<<<NP, 128, 0, stream>>>(states, users, items, W1, b1, W2, b2, out);
}